// SelectiveScanCore_78323023610178
// MI455X (gfx1250) — compile-verified
//
#include <hip/hip_runtime.h>
#include <hip/hip_bf16.h>

typedef float v2f __attribute__((ext_vector_type(2)));
typedef float v8f __attribute__((ext_vector_type(8)));

#define D_MODEL 1024
#define D_STATE 16
#define DT_RANK 64
#define BATCH   4
#define SEQLEN  2048
#define XDIM    96                 // DT_RANK + 2*D_STATE
#define NTOK    (BATCH * SEQLEN)   // 8192 tokens
#define NCHUNK  32
#define CLEN    (SEQLEN / NCHUNK)  // 64 steps per chunk
#define NT1     (XDIM / 16)        // 6 n-tiles per wave (xp kernel)
#define NT2     8                  // n-tiles per wave (delta kernel)

// ---------------------------------------------------------------------------
// Kernel 1: xp[m, 0:96] = sum_k u[m,k] * x_proj_w[n,k]   (f32 WMMA 16x16x4)
// One wave owns a 16-row stripe and ALL 6 n-tiles: A (u) is loaded once per
// k-step and shared across 6 independent accumulators -> 6x less A traffic
// and no serial WMMA D->C chain. Grid exact -> EXEC all ones.
// A 16x4 f32 layout: lane<16 holds K=0 (v0), K=1 (v1); lane>=16 holds K=2,3.
// ---------------------------------------------------------------------------
__global__ __launch_bounds__(256) void xp_wmma_kernel(const float* __restrict__ u,
                                                      const float* __restrict__ xw,
                                                      float* __restrict__ xp) {
  const int gtid   = blockIdx.x * 256 + threadIdx.x;
  const int m_tile = gtid >> 5;                    // 512 waves = 512 m-tiles
  const int lane   = threadIdx.x & 31;
  const int row    = lane & 15;
  const int kph    = (lane >> 4) << 1;             // 0 or 2

  const float* __restrict__ ap = u  + (size_t)(m_tile * 16 + row) * D_MODEL + kph;
  const float* __restrict__ bp = xw + (size_t)row * D_MODEL + kph;

  v8f acc[NT1] = {};
  for (int k = 0; k < D_MODEL; k += 4) {
    v2f a = *(const v2f*)(ap + k);
    #pragma unroll
    for (int nt = 0; nt < NT1; ++nt) {
      v2f b = *(const v2f*)(bp + (size_t)nt * 16 * D_MODEL + k);
      acc[nt] = __builtin_amdgcn_wmma_f32_16x16x4_f32(false, a, false, b, (short)0,
                                                      acc[nt], false, false);
    }
  }

  const int m0  = m_tile * 16 + ((lane >> 4) << 3);
  const int col = lane & 15;
  #pragma unroll
  for (int nt = 0; nt < NT1; ++nt)
    #pragma unroll
    for (int i = 0; i < 8; ++i)
      xp[(size_t)(m0 + i) * XDIM + nt * 16 + col] = acc[nt][i];
}

// ---------------------------------------------------------------------------
// Kernel 2: delta[m,n] = softplus( sum_k xp[m,k<64] * dt_proj_w[n,k] + b[n] )
// One wave owns a 16-row stripe and 8 n-tiles (8 independent accumulators).
// ---------------------------------------------------------------------------
__global__ __launch_bounds__(256) void delta_wmma_kernel(const float* __restrict__ xp,
                                                         const float* __restrict__ wt,
                                                         const float* __restrict__ bias,
                                                         float* __restrict__ delta) {
  const int gtid   = blockIdx.x * 256 + threadIdx.x;
  const int wave   = gtid >> 5;                    // 4096 waves
  const int lane   = threadIdx.x & 31;
  const int ngrp   = wave % (D_MODEL / 16 / NT2);  // 0..7
  const int m_tile = wave / (D_MODEL / 16 / NT2);  // 0..511
  const int row    = lane & 15;
  const int kph    = (lane >> 4) << 1;

  const float* __restrict__ ap = xp + (size_t)(m_tile * 16 + row) * XDIM + kph;
  const float* __restrict__ bp = wt + (size_t)(ngrp * NT2 * 16 + row) * DT_RANK + kph;

  v8f acc[NT2] = {};
  #pragma unroll
  for (int k = 0; k < DT_RANK; k += 4) {
    v2f a = *(const v2f*)(ap + k);
    #pragma unroll
    for (int nt = 0; nt < NT2; ++nt) {
      v2f b = *(const v2f*)(bp + (size_t)nt * 16 * DT_RANK + k);
      acc[nt] = __builtin_amdgcn_wmma_f32_16x16x4_f32(false, a, false, b, (short)0,
                                                      acc[nt], false, false);
    }
  }

  const int m0 = m_tile * 16 + ((lane >> 4) << 3);
  #pragma unroll
  for (int nt = 0; nt < NT2; ++nt) {
    const int n0 = (ngrp * NT2 + nt) * 16 + (lane & 15);
    const float bn = bias[n0];
    #pragma unroll
    for (int i = 0; i < 8; ++i) {
      float x  = acc[nt][i] + bn;
      float sp = (x > 20.0f) ? x : __logf(1.0f + __expf(x));
      delta[(size_t)(m0 + i) * D_MODEL + n0] = sp;
    }
  }
}

// ---------------------------------------------------------------------------
// Pass 1 of chunked scan: per (b, chunk, d) compute zero-init chunk result
// Q[16] and sum of dt (chunk transfer matrix is exp(A * Sdt)).
// b, c derived from blockIdx.x only -> B_t addresses provably wave-uniform
// so the float4 broadcasts can lower to scalar (SMEM) loads.
// Grid: BATCH * NCHUNK * (D_MODEL/256) = 512 blocks.
// ---------------------------------------------------------------------------
__global__ __launch_bounds__(256) void scan_pass1_kernel(const float* __restrict__ delta,
                                                         const float* __restrict__ xp,
                                                         const float* __restrict__ u,
                                                         const float* __restrict__ A_log,
                                                         float* __restrict__ Q,
                                                         float* __restrict__ Sdt) {
  const int bid = blockIdx.x;
  const int d   = (bid & 3) * 256 + threadIdx.x;   // D_MODEL/256 = 4 blocks per (b,c)
  const int c   = (bid >> 2) & (NCHUNK - 1);
  const int b   = bid >> 7;

  float A[D_STATE];
  #pragma unroll
  for (int n = 0; n < D_STATE; ++n) A[n] = -__expf(A_log[d * D_STATE + n]);

  float h[D_STATE];
  #pragma unroll
  for (int n = 0; n < D_STATE; ++n) h[n] = 0.0f;
  float sdt = 0.0f;

  const size_t tok0 = (size_t)b * SEQLEN + (size_t)c * CLEN;   // scalar
  for (int s = 0; s < CLEN; ++s) {
    const size_t tok = tok0 + s;                               // scalar
    const float dt = delta[tok * D_MODEL + d];
    const float ut = u[tok * D_MODEL + d];
    const float4* bv = (const float4*)(xp + tok * XDIM + DT_RANK);  // uniform addr
    float4 q0 = bv[0], q1 = bv[1], q2 = bv[2], q3 = bv[3];
    float Bt[D_STATE] = {q0.x, q0.y, q0.z, q0.w, q1.x, q1.y, q1.z, q1.w,
                         q2.x, q2.y, q2.z, q2.w, q3.x, q3.y, q3.z, q3.w};
    sdt += dt;
    const float du = dt * ut;
    #pragma unroll
    for (int n = 0; n < D_STATE; ++n)
      h[n] = __expf(dt * A[n]) * h[n] + du * Bt[n];
  }

  const size_t idx = ((size_t)b * NCHUNK + c) * D_MODEL + d;
  #pragma unroll
  for (int n = 0; n < D_STATE; ++n) Q[idx * D_STATE + n] = h[n];
  Sdt[idx] = sdt;
}

// ---------------------------------------------------------------------------
// Carry pass: per (b, d), sequentially combine 32 chunk summaries.
// Hc[b, c, d, :] = carry-in state for chunk c.
// ---------------------------------------------------------------------------
__global__ __launch_bounds__(256) void scan_carry_kernel(const float* __restrict__ Q,
                                                         const float* __restrict__ Sdt,
                                                         const float* __restrict__ A_log,
                                                         float* __restrict__ Hc) {
  const int tid = blockIdx.x * 256 + threadIdx.x;   // B*D_MODEL threads
  const int d   = tid % D_MODEL;
  const int b   = tid / D_MODEL;

  float A[D_STATE];
  #pragma unroll
  for (int n = 0; n < D_STATE; ++n) A[n] = -__expf(A_log[d * D_STATE + n]);

  float h[D_STATE];
  #pragma unroll
  for (int n = 0; n < D_STATE; ++n) h[n] = 0.0f;

  for (int c = 0; c < NCHUNK; ++c) {
    const size_t idx = ((size_t)b * NCHUNK + c) * D_MODEL + d;
    #pragma unroll
    for (int n = 0; n < D_STATE; ++n) Hc[idx * D_STATE + n] = h[n];
    const float s = Sdt[idx];
    #pragma unroll
    for (int n = 0; n < D_STATE; ++n)
      h[n] = __expf(s * A[n]) * h[n] + Q[idx * D_STATE + n];
  }
}

// ---------------------------------------------------------------------------
// Pass 3: replay each chunk from its carry-in, emit y = sum_n h*C + u*D_skip.
// Same uniform-address trick for B_t / C_t broadcasts.
// ---------------------------------------------------------------------------
__global__ __launch_bounds__(256) void scan_pass3_kernel(const float* __restrict__ delta,
                                                         const float* __restrict__ xp,
                                                         const float* __restrict__ u,
                                                         const float* __restrict__ A_log,
                                                         const float* __restrict__ Dsk,
                                                         const float* __restrict__ Hc,
                                                         float* __restrict__ y) {
  const int bid = blockIdx.x;
  const int d   = (bid & 3) * 256 + threadIdx.x;
  const int c   = (bid >> 2) & (NCHUNK - 1);
  const int b   = bid >> 7;

  float A[D_STATE];
  #pragma unroll
  for (int n = 0; n < D_STATE; ++n) A[n] = -__expf(A_log[d * D_STATE + n]);

  const size_t idx = ((size_t)b * NCHUNK + c) * D_MODEL + d;
  float h[D_STATE];
  #pragma unroll
  for (int n = 0; n < D_STATE; ++n) h[n] = Hc[idx * D_STATE + n];
  const float dsk = Dsk[d];

  const size_t tok0 = (size_t)b * SEQLEN + (size_t)c * CLEN;
  for (int s = 0; s < CLEN; ++s) {
    const size_t tok = tok0 + s;
    const float dt = delta[tok * D_MODEL + d];
    const float ut = u[tok * D_MODEL + d];
    const float4* bv = (const float4*)(xp + tok * XDIM + DT_RANK);
    float4 q0 = bv[0], q1 = bv[1], q2 = bv[2], q3 = bv[3];
    const float4* cv = (const float4*)(xp + tok * XDIM + DT_RANK + D_STATE);
    float4 p0 = cv[0], p1 = cv[1], p2 = cv[2], p3 = cv[3];
    float Bt[D_STATE] = {q0.x, q0.y, q0.z, q0.w, q1.x, q1.y, q1.z, q1.w,
                         q2.x, q2.y, q2.z, q2.w, q3.x, q3.y, q3.z, q3.w};
    float Ct[D_STATE] = {p0.x, p0.y, p0.z, p0.w, p1.x, p1.y, p1.z, p1.w,
                         p2.x, p2.y, p2.z, p2.w, p3.x, p3.y, p3.z, p3.w};
    const float du = dt * ut;
    float acc = 0.0f;
    #pragma unroll
    for (int n = 0; n < D_STATE; ++n) {
      h[n] = __expf(dt * A[n]) * h[n] + du * Bt[n];
      acc += h[n] * Ct[n];
    }
    y[tok * D_MODEL + d] = acc + ut * dsk;
  }
}

// ---------------------------------------------------------------------------
extern "C" void kernel_launch(void* const* d_in, const int* in_sizes, int n_in,
                              void* d_out, int out_size, void* d_ws, size_t ws_size,
                              hipStream_t stream) {
  const float* u      = (const float*)d_in[0];  // (4, 2048, 1024)
  const float* A_log  = (const float*)d_in[1];  // (1024, 16)
  const float* D_skip = (const float*)d_in[2];  // (1024,)
  const float* dt_w   = (const float*)d_in[3];  // (1024, 64)
  const float* dt_b   = (const float*)d_in[4];  // (1024,)
  const float* x_w    = (const float*)d_in[5];  // (96, 1024)
  float* y = (float*)d_out;

  // workspace carve-up (all offsets 16B-aligned)
  float* ws    = (float*)d_ws;
  float* xp    = ws;                                       // 8192*96
  float* delta = xp    + (size_t)NTOK * XDIM;              // 8192*1024
  float* Q     = delta + (size_t)NTOK * D_MODEL;           // 131072*16
  float* Sdt   = Q     + (size_t)BATCH * NCHUNK * D_MODEL * D_STATE; // 131072
  float* Hc    = Sdt   + (size_t)BATCH * NCHUNK * D_MODEL;           // 131072*16

  // 1) xp projection: 512 waves (one per 16-row stripe, 6 n-tiles each)
  xp_wmma_kernel<<<(NTOK / 16) * 32 / 256, 256, 0, stream>>>(u, x_w, xp);

  // 2) delta projection + softplus: 512 stripes x 8 n-groups = 4096 waves
  delta_wmma_kernel<<<(NTOK / 16) * (D_MODEL / 16 / NT2) * 32 / 256, 256, 0, stream>>>(
      xp, dt_w, dt_b, delta);

  // 3) chunked scan: 4*32*1024 = 131072 threads, (b,c) uniform per block
  scan_pass1_kernel<<<BATCH * NCHUNK * (D_MODEL / 256), 256, 0, stream>>>(
      delta, xp, u, A_log, Q, Sdt);

  // 4) sequential carry across chunks: 4096 threads
  scan_carry_kernel<<<BATCH * D_MODEL / 256, 256, 0, stream>>>(Q, Sdt, A_log, Hc);

  // 5) replay + output
  scan_pass3_kernel<<<BATCH * NCHUNK * (D_MODEL / 256), 256, 0, stream>>>(
      delta, xp, u, A_log, D_skip, Hc, y);
}